// BinnedSpectraLayer_42545946034791
// MI455X (gfx1250) — compile-verified
//
#include <hip/hip_runtime.h>
#include <cstdint>

// BinnedSpectraLayer for MI455X (gfx1250, wave32).
//
// Strategy: one 256-thread workgroup per spectrum row.
//   phase 1: zero a 9900-float LDS histogram (float4 stores)
//   phase 2: each thread takes 4 peaks (one float4 of mz + one of intensity),
//            ds_add_f32 LDS atomics into the histogram
//   phase 3: flush LDS row -> global with global_store_async_from_lds_b128
//            (CDNA5 async LDS->memory DMA, ASYNCcnt), then s_wait_asynccnt 0.
//
// The kernel is output-bandwidth bound (~162 MB written); the async path
// avoids the ds_load+global_store round trip per 16B chunk.

namespace {

constexpr float kMinMz       = 10.0f;
constexpr float kMaxMz       = 1000.0f;
constexpr float kBinWidth    = 0.1f;
constexpr int   kNumBins     = 9900;   // (1000 - 10) / 0.1
constexpr int   kPeaks       = 1024;   // P
constexpr int   kThreads     = 256;    // 8 wave32 per block
constexpr int   kVec         = kNumBins / 4;  // 2475 float4 chunks (exact)

} // namespace

__global__ __launch_bounds__(kThreads)
void binned_spectra_kernel(const float* __restrict__ mz,
                           const float* __restrict__ inten,
                           float* __restrict__ out)
{
    __shared__ __align__(16) float hist[kNumBins];   // 39600 B LDS

    const int row = blockIdx.x;
    const int tid = threadIdx.x;

    // ---- phase 1: zero histogram ---------------------------------------
    float4* h4 = reinterpret_cast<float4*>(hist);
    for (int i = tid; i < kVec; i += kThreads) {
        h4[i] = make_float4(0.0f, 0.0f, 0.0f, 0.0f);
    }
    __syncthreads();

    // ---- phase 2: scatter-add peaks (4 per thread, vectorized loads) ----
    const float4* mz4 = reinterpret_cast<const float4*>(mz    + (size_t)row * kPeaks);
    const float4* iv4 = reinterpret_cast<const float4*>(inten + (size_t)row * kPeaks);
    // kPeaks / 4 == kThreads: exactly one float4 pair per thread.
    const float4 m = mz4[tid];
    const float4 v = iv4[tid];

    const float mzs[4] = { m.x, m.y, m.z, m.w };
    const float ivs[4] = { v.x, v.y, v.z, v.w };
    #pragma unroll
    for (int k = 0; k < 4; ++k) {
        const float mzv = mzs[k];
        if (mzv >= kMinMz && mzv < kMaxMz) {
            int bin = (int)((mzv - kMinMz) / kBinWidth);   // matches reference math
            bin = bin < 0 ? 0 : (bin > kNumBins - 1 ? kNumBins - 1 : bin);
            atomicAdd(&hist[bin], sqrtf(ivs[k]));          // ds_add_f32
        }
    }
    __syncthreads();   // all LDS atomics visible before async engine reads LDS

    // ---- phase 3: async DMA LDS -> global (16B per instruction) ---------
    float* orow = out + (size_t)row * kNumBins;  // 39600B row stride, 16B aligned
    for (int i = tid; i < kVec; i += kThreads) {
        void*    gaddr = (void*)(orow + 4 * i);
        unsigned laddr = (unsigned)(uintptr_t)(&hist[4 * i]);  // low 32b = LDS offset
        asm volatile("global_store_async_from_lds_b128 %0, %1, off"
                     :
                     : "v"(gaddr), "v"(laddr)
                     : "memory");
    }
#if __has_builtin(__builtin_amdgcn_s_wait_asynccnt)
    __builtin_amdgcn_s_wait_asynccnt(0);
#else
    asm volatile("s_wait_asynccnt 0x0" ::: "memory");
#endif
}

extern "C" void kernel_launch(void* const* d_in, const int* in_sizes, int n_in,
                              void* d_out, int out_size, void* d_ws, size_t ws_size,
                              hipStream_t stream)
{
    (void)n_in; (void)out_size; (void)d_ws; (void)ws_size;

    const float* mz    = (const float*)d_in[0];   // [B, 1024] f32
    const float* inten = (const float*)d_in[1];   // [B, 1024] f32
    float*       out   = (float*)d_out;           // [B, 9900] f32

    const int rows = in_sizes[0] / kPeaks;        // B = 4096

    binned_spectra_kernel<<<rows, kThreads, 0, stream>>>(mz, inten, out);
}